// Regrid_84378927497346
// MI455X (gfx1250) — compile-verified
//
#include <hip/hip_runtime.h>

// ---------------------------------------------------------------------------
// ERA5 (721x1440) -> CESM (192x288) sparse bilinear regrid, M=120 channels.
//   out[m, r] += w[i] * x[m, flip_lat(col[i])]  for each nnz i with row[i]==r
//
// Memory-bound: ~498 MB gather reads + 26.5 MB output, ~53 MFLOP.
// Floor at 23.3 TB/s ~= 22.5 us. WMMA is inapplicable (per-row scatter
// columns defeat the shared-K contraction); optimize the memory path:
//  - nnz-fastest dispatch keeps the 8 active x channel-slabs (~33 MB)
//    L2-resident (192 MB), so each 128B line is fetched from HBM once.
//  - gathers ride the gfx1250 async global->LDS path (ASYNCcnt), software-
//    pipelined across tiles with s_wait_asynccnt 8 (in-order completion).
//  - scatter uses hardware global_atomic_add_f32; the live output slice
//    (8 x 221 KB) stays in L2.
// ---------------------------------------------------------------------------

constexpr int NLON = 1440;
constexpr int NLAT = 721;
constexpr int NA_I = NLAT * NLON;   // 1,038,240 source cells
constexpr int NB_I = 192 * 288;     //    55,296 destination cells
constexpr int MCHUNK = 8;           // channel slices per block (120 = 15 * 8)
constexpr int TPB = 256;            // 8 waves (wave32)
constexpr int TILES_PER_BLOCK = 4;  // software-pipelined nnz tiles per block

// gfx1250 async global->LDS gather path, if this toolchain exposes it.
#if defined(__HIP_DEVICE_COMPILE__)
#if __has_builtin(__builtin_amdgcn_global_load_async_to_lds_b32) && \
    __has_builtin(__builtin_amdgcn_s_wait_asynccnt)
#define USE_ASYNC_LDS 1
#endif
#endif
#ifndef USE_ASYNC_LDS
#define USE_ASYNC_LDS 0
#endif

#if USE_ASYNC_LDS
typedef __attribute__((address_space(1))) int GInt;  // global src (per hipcc sig)
typedef __attribute__((address_space(3))) int LInt;  // LDS dst
#endif

__global__ __launch_bounds__(TPB) void regrid_zero_f32(float* __restrict__ p, int n) {
  const int i = blockIdx.x * TPB + threadIdx.x;
  const int i4 = i * 4;
  if (i4 + 3 < n) {
    *reinterpret_cast<float4*>(p + i4) = make_float4(0.f, 0.f, 0.f, 0.f);
  } else {
    for (int k = i4; k < n; ++k) p[k] = 0.0f;
  }
}

__global__ __launch_bounds__(TPB) void regrid_scatter(
    const float* __restrict__ x,    // [M, 721, 1440]
    const float* __restrict__ wgt,  // [nnz]
    const int* __restrict__ row,    // [nnz] in [0, NB)
    const int* __restrict__ col,    // [nnz] in [0, NA), indexes lat-flipped x
    float* __restrict__ out,        // [M, NB]
    int nnz, int M) {
  const int ntiles = (nnz + TPB - 1) / TPB;
  const int tile0  = blockIdx.x * TILES_PER_BLOCK;
  const int m0     = blockIdx.y * MCHUNK;
  const int t      = threadIdx.x;
  const int nt     = min(TILES_PER_BLOCK, ntiles - tile0);
  if (nt <= 0) return;

#if USE_ASYNC_LDS
  // Double-buffered stage: lane t owns stage[b][j*TPB + t] (conflict-free,
  // 4B lane stride). 2 * 8 * 256 * 4B = 16 KB of the 320 KB WGP LDS.
  __shared__ float stage[2][MCHUNK * TPB];
  float wbuf[2];
  int   rbuf[2];

  // Issue exactly MCHUNK async gathers for tile k into buffer b. The count
  // is EXEC/guard-invariant (clamped index + clamped channel offset), so a
  // later s_wait_asynccnt MCHUNK precisely retires this tile's loads
  // (ASYNCcnt loads complete in order).
  auto issue_tile = [&](int k, int b) {
    const int idx = (tile0 + k) * TPB + t;
    const int ic  = idx < nnz ? idx : nnz - 1;   // clamp, weight zeroed below
    wbuf[b] = idx < nnz ? wgt[ic] : 0.0f;
    rbuf[b] = row[ic];
    const int a   = col[ic];
    const int lat = a / NLON;                    // col indexes lat-flipped x
    const int lon = a - lat * NLON;
    const long long src = (long long)(NLAT - 1 - lat) * NLON + lon;
    const float* xp = x + (long long)m0 * NA_I + src;
#pragma unroll
    for (int j = 0; j < MCHUNK; ++j) {
      const long long off = (m0 + j < M) ? (long long)j * NA_I : 0;  // clamp
      __builtin_amdgcn_global_load_async_to_lds_b32(
          (GInt*)(xp + off), (LInt*)&stage[b][j * TPB + t], 0, 0);
    }
  };

  issue_tile(0, 0);
#pragma unroll
  for (int k = 0; k < TILES_PER_BLOCK; ++k) {
    if (k >= nt) break;
    const int b = k & 1;
    if (k + 1 < nt) {
      issue_tile(k + 1, (k + 1) & 1);            // prefetch next tile
      __builtin_amdgcn_s_wait_asynccnt(MCHUNK);  // tile k landed; k+1 in flight
    } else {
      __builtin_amdgcn_s_wait_asynccnt(0);
    }
    const float wv = wbuf[b];
    float* op = out + (long long)m0 * NB_I + rbuf[b];
#pragma unroll
    for (int j = 0; j < MCHUNK; ++j) {
      if (m0 + j < M) {
        __hip_atomic_fetch_add(op + (long long)j * NB_I,
                               wv * stage[b][j * TPB + t],
                               __ATOMIC_RELAXED, __HIP_MEMORY_SCOPE_AGENT);
      }
    }
  }
#else
  // Fallback: direct gathers with prefetch + MLP across the 8 channel slabs.
  for (int k = 0; k < nt; ++k) {
    const int idx = (tile0 + k) * TPB + t;
    if (idx >= nnz) break;
    const int a   = col[idx];
    const int r   = row[idx];
    const float w = wgt[idx];
    const int lat = a / NLON;
    const int lon = a - lat * NLON;
    const long long src = (long long)(NLAT - 1 - lat) * NLON + lon;
    const float* xp = x + (long long)m0 * NA_I + src;
    float* op = out + (long long)m0 * NB_I + r;
    float v[MCHUNK];
#pragma unroll
    for (int j = 0; j < MCHUNK; ++j) {
      if (m0 + j < M) __builtin_prefetch(xp + (long long)j * NA_I, 0, 1);
    }
#pragma unroll
    for (int j = 0; j < MCHUNK; ++j) {
      v[j] = (m0 + j < M) ? xp[(long long)j * NA_I] : 0.0f;
    }
#pragma unroll
    for (int j = 0; j < MCHUNK; ++j) {
      if (m0 + j < M) {
        __hip_atomic_fetch_add(op + (long long)j * NB_I, w * v[j],
                               __ATOMIC_RELAXED, __HIP_MEMORY_SCOPE_AGENT);
      }
    }
  }
#endif
}

extern "C" void kernel_launch(void* const* d_in, const int* in_sizes, int n_in,
                              void* d_out, int out_size, void* d_ws, size_t ws_size,
                              hipStream_t stream) {
  const float* x   = (const float*)d_in[0];
  const float* w   = (const float*)d_in[1];
  const int*   row = (const int*)d_in[2];
  const int*   col = (const int*)d_in[3];
  float*       out = (float*)d_out;

  const int nnz = in_sizes[1];              // 221,184
  const int M   = in_sizes[0] / NA_I;       // 120

  // Harness poisons d_out; zero it first (same stream -> ordered).
  const int zthreads = (out_size + 3) / 4;
  hipLaunchKernelGGL(regrid_zero_f32, dim3((zthreads + TPB - 1) / TPB), dim3(TPB),
                     0, stream, out, out_size);

  // x-dim (nnz tiles) varies fastest in dispatch order -> all blocks of one
  // channel chunk run together, keeping its x-slabs L2-resident.
  const int ntiles = (nnz + TPB - 1) / TPB;                    // 864
  dim3 grid((ntiles + TILES_PER_BLOCK - 1) / TILES_PER_BLOCK,  // 216
            (M + MCHUNK - 1) / MCHUNK);                        // 15
  hipLaunchKernelGGL(regrid_scatter, grid, dim3(TPB), 0, stream,
                     x, w, row, col, out, nnz, M);
}